// DynamicNTXentLoss_59330678227334
// MI455X (gfx1250) — compile-verified
//
#include <hip/hip_runtime.h>
#include <math.h>

// NT-Xent loss, memory-bound streaming kernel using V_WMMA_F32_16X16X4_F32
// to build per-group 4x4 Gram blocks (4 groups / 16 rows per wave, K=256).

typedef __attribute__((ext_vector_type(2))) float v2f;
typedef __attribute__((ext_vector_type(8))) float v8f;

#define NTX_EPS 1e-8f

__device__ __forceinline__ float pick4(int j, float a0, float a1, float a2, float a3) {
    return (j == 0) ? a0 : (j == 1) ? a1 : (j == 2) ? a2 : a3;
}

__global__ __launch_bounds__(256) void ntxent_gram_kernel(
    const float* __restrict__ zis, const float* __restrict__ zjs,
    const float* __restrict__ soft_labels, float* __restrict__ partials)
{
    const int lane = threadIdx.x & 31;
    const int wave = threadIdx.x >> 5;
    const long tile = (long)blockIdx.x * 8 + wave;   // 4 groups (16 rows) per wave

    // Which of the 16 tile rows this lane feeds into the WMMA A/B operands.
    // Row m of tile: group g = m>>2, within-group index i = m&3.
    // Group rows are [zj[2c], zj[2c+1], zi[2c], zi[2c+1]].
    const int m = lane & 15;
    const int g = m >> 2;
    const int i = m & 3;
    const long grp = tile * 4 + g;
    const float* src = (i < 2) ? zjs : zis;
    // lanes 0-15 supply K pair (4k, 4k+1); lanes 16-31 supply (4k+2, 4k+3)
    const float* p = src + (grp * 2 + (long)(i & 1)) * 256 + ((lane >> 4) << 1);

    // Accumulate S = R * R^T (16x16) over K=256 in 64 steps of K=4.
    // For this symmetric product the A and B register images are identical,
    // so one float2 load per lane per step feeds both operands.
    v8f acc = {};
    #pragma unroll
    for (int k = 0; k < 64; ++k) {
        v2f a = *(const v2f*)(p + 4 * k);
        acc = __builtin_amdgcn_wmma_f32_16x16x4_f32(
            /*neg_a=*/false, a, /*neg_b=*/false, a,
            /*c_mod=*/(short)0, acc, /*reuse_a=*/false, /*reuse_b=*/false);
    }

    // C layout: lane L<16 holds column N=L rows M=0..7 in acc[0..7];
    // lane L>=16 holds column N=L-16 rows M=8..15 in acc[0..7].
    // The 4x4 diagonal blocks therefore live in lanes 0-7 and 24-31:
    //   g0 -> lanes 0-3  acc[0..3], g1 -> lanes 4-7  acc[4..7],
    //   g2 -> lanes 24-27 acc[0..3], g3 -> lanes 28-31 acc[4..7].
    // Each such lane holds one full row (== column, symmetric) of its block.
    const int gg = (lane < 16) ? (lane >> 2) : ((lane - 16) >> 2);
    const bool hi = (gg & 1) != 0;
    float w0 = hi ? acc[4] : acc[0];
    float w1 = hi ? acc[5] : acc[1];
    float w2 = hi ? acc[6] : acc[2];
    float w3 = hi ? acc[7] : acc[3];

    const int j = lane & 3;                 // row index within the 4x4 block
    float dj = pick4(j, w0, w1, w2, w3);    // own diagonal S[j][j]

    const int base = lane & ~3;             // quad-aligned shuffle base
    float d0 = __shfl(dj, base + 0, 32);
    float d1 = __shfl(dj, base + 1, 32);
    float d2 = __shfl(dj, base + 2, 32);
    float d3 = __shfl(dj, base + 3, 32);

    float n0 = fmaxf(sqrtf(d0), NTX_EPS);
    float n1 = fmaxf(sqrtf(d1), NTX_EPS);
    float n2 = fmaxf(sqrtf(d2), NTX_EPS);
    float n3 = fmaxf(sqrtf(d3), NTX_EPS);
    float nj = pick4(j, n0, n1, n2, n3);

    float s0 = w0 / (nj * n0);
    float s1 = w1 / (nj * n1);
    float s2 = w2 / (nj * n2);
    float s3 = w3 / (nj * n3);

    // logits / T, T = 0.5: pos = sim[j][j^2], negs = sim[j][j^1], sim[j][j^3]
    float lp = 2.0f * pick4(j ^ 2, s0, s1, s2, s3);
    float la = 2.0f * pick4(j ^ 1, s0, s1, s2, s3);
    float lb = 2.0f * pick4(j ^ 3, s0, s1, s2, s3);

    float mx  = fmaxf(lp, fmaxf(la, lb));
    float lse = mx + logf(expf(lp - mx) + expf(la - mx) + expf(lb - mx));

    float sl0 = soft_labels[j * 3 + 0];
    float sl1 = soft_labels[j * 3 + 1];
    float sl2 = soft_labels[j * 3 + 2];
    float contrib = -(sl0 * (lp - lse) + sl1 * (la - lse) + sl2 * (lb - lse));

    const bool active = (lane < 8) || (lane >= 24);
    contrib = active ? contrib : 0.0f;

    // wave32 reduction
    #pragma unroll
    for (int o = 16; o > 0; o >>= 1)
        contrib += __shfl_xor(contrib, o, 32);

    __shared__ float smem[8];
    if (lane == 0) smem[wave] = contrib;
    __syncthreads();
    if (threadIdx.x == 0) {
        float t = 0.0f;
        #pragma unroll
        for (int w = 0; w < 8; ++w) t += smem[w];
        partials[blockIdx.x] = t;
    }
}

__global__ __launch_bounds__(256) void ntxent_finalize_kernel(
    const float* __restrict__ partials, int nparts, float* __restrict__ out, float scale)
{
    __shared__ float smem[256];
    float t = 0.0f;
    for (int idx = threadIdx.x; idx < nparts; idx += 256) t += partials[idx];
    smem[threadIdx.x] = t;
    __syncthreads();
    for (int s = 128; s > 0; s >>= 1) {
        if (threadIdx.x < s) smem[threadIdx.x] += smem[threadIdx.x + s];
        __syncthreads();
    }
    if (threadIdx.x == 0) out[0] = smem[0] * scale;
}

extern "C" void kernel_launch(void* const* d_in, const int* in_sizes, int n_in,
                              void* d_out, int out_size, void* d_ws, size_t ws_size,
                              hipStream_t stream) {
    const float* zis  = (const float*)d_in[0];
    const float* zjs  = (const float*)d_in[1];
    const float* soft = (const float*)d_in[2];
    float* out      = (float*)d_out;
    float* partials = (float*)d_ws;

    const int  D     = 256;
    const long nrows = (long)in_sizes[0] / D;  // 65536
    const long nc    = nrows / 2;              // 32768 groups
    const long tiles = nc / 4;                 // 8192 waves
    const int  blocks = (int)(tiles / 8);      // 1024 blocks of 8 waves

    ntxent_gram_kernel<<<blocks, 256, 0, stream>>>(zis, zjs, soft, partials);

    const float scale = 1.0f / (float)(2 * nrows);
    ntxent_finalize_kernel<<<1, 256, 0, stream>>>(partials, blocks, out, scale);
}